// GCNEncoder_18897856102726
// MI455X (gfx1250) — compile-verified
//
#include <hip/hip_runtime.h>
#include <hip/hip_bf16.h>
#include <stdint.h>

typedef __attribute__((ext_vector_type(2))) float v2f;
typedef __attribute__((ext_vector_type(8))) float v8f;

#define CH 64           // HID == OUT_CH == 64
#define WAVES_PER_BLOCK 8

// ---------------------------------------------------------------------------
// Elementwise / degree kernels
// ---------------------------------------------------------------------------
__global__ void fill_f32(float* __restrict__ p, float v, long long n) {
    long long i = blockIdx.x * (long long)blockDim.x + threadIdx.x;
    if (i < n) p[i] = v;
}

__global__ void deg_accum(const long long* __restrict__ dst, float* deg, long long E) {
    long long e = blockIdx.x * (long long)blockDim.x + threadIdx.x;
    if (e < E) atomicAdd(&deg[dst[e]], 1.0f);
}

__global__ void rsqrt_inplace(float* __restrict__ p, long long n) {
    long long i = blockIdx.x * (long long)blockDim.x + threadIdx.x;
    if (i < n) p[i] = rsqrtf(p[i]);
}

// agg[i,c] = h[i,c] * dinv[i]^2   (self-loop message)
__global__ void self_loop_init(const float* __restrict__ h, const float* __restrict__ dinv,
                               float* __restrict__ agg, long long total) {
    long long i = blockIdx.x * (long long)blockDim.x + threadIdx.x;
    if (i >= total) return;
    float di = dinv[i >> 6];          // node = i / 64
    agg[i] = h[i] * di * di;
}

// scatter: agg[dst,c] += h[src,c] * dinv[src]*dinv[dst]; 16 threads/edge, 4 ch each
__global__ void edge_scatter(const long long* __restrict__ src, const long long* __restrict__ dst,
                             const float* __restrict__ dinv, const float* __restrict__ h,
                             float* agg, long long E) {
    long long t = blockIdx.x * (long long)blockDim.x + threadIdx.x;
    long long e = t >> 4;
    if (e >= E) return;
    int c = (int)(t & 15) * 4;
    long long s = src[e], d = dst[e];
    float coef = dinv[s] * dinv[d];
    const float* hp = h + s * CH + c;
    float* ap = agg + d * CH + c;
    atomicAdd(ap + 0, hp[0] * coef);
    atomicAdd(ap + 1, hp[1] * coef);
    atomicAdd(ap + 2, hp[2] * coef);
    atomicAdd(ap + 3, hp[3] * coef);
}

__global__ void bias_act_inplace(float* __restrict__ buf, const float* __restrict__ bias,
                                 long long total, int do_relu) {
    long long i = blockIdx.x * (long long)blockDim.x + threadIdx.x;
    if (i >= total) return;
    float v = buf[i] + bias[i & (CH - 1)];
    if (do_relu) v = fmaxf(v, 0.0f);
    buf[i] = v;
}

__global__ void add_bias_store(const float* __restrict__ in, const float* __restrict__ bias,
                               float* __restrict__ out, long long total) {
    long long i = blockIdx.x * (long long)blockDim.x + threadIdx.x;
    if (i >= total) return;
    out[i] = in[i] + bias[i & (CH - 1)];
}

// ---------------------------------------------------------------------------
// fp32 WMMA GEMM: C[nrows x 64] = A[nrows x K] * W[K x 64]
// One wave32 computes one 16x16 tile via V_WMMA_F32_16X16X4_F32, K stepped by 4.
// ---------------------------------------------------------------------------
template <int K>
__global__ __launch_bounds__(WAVES_PER_BLOCK * 32)
void gemm_wmma_f32(const float* __restrict__ A, const float* __restrict__ W,
                   float* __restrict__ C, int nrows) {
    const int tilesN = CH / 16;                       // 4 column tiles
    int wave = threadIdx.x >> 5;
    int lane = threadIdx.x & 31;
    int tile = blockIdx.x * WAVES_PER_BLOCK + wave;
    int nTiles = (nrows / 16) * tilesN;
    if (tile >= nTiles) return;                       // wave-uniform; EXEC stays full

    int tm = tile / tilesN, tn = tile % tilesN;
    int row0 = tm * 16, col0 = tn * 16;
    int half = lane >> 4;                             // lane half selects K sub-pair
    int l16  = lane & 15;

    v8f acc = {};
    // A frag: lane half h -> K = k + 2h + {0,1}, row = row0 + l16
    const float* ap = A + (size_t)(row0 + l16) * K + 2 * half;
    // B frag: lane half h -> K = k + 2h + vgpr, col = col0 + l16
    const float* bp = W + (size_t)(2 * half) * CH + col0 + l16;

    for (int k = 0; k < K; k += 4) {
        v2f a, b;
        a.x = ap[0];
        a.y = ap[1];
        b.x = bp[0];
        b.y = bp[CH];
        acc = __builtin_amdgcn_wmma_f32_16x16x4_f32(
            /*neg_a=*/false, a, /*neg_b=*/false, b,
            /*c_mod=*/(short)0, acc, /*reuse_a=*/false, /*reuse_b=*/false);
        ap += 4;
        bp += 4 * CH;
    }

    // C/D layout: VGPR r -> row = row0 + r + 8*half, col = col0 + l16
    float* cp = C + (size_t)(row0 + 8 * half) * CH + col0 + l16;
#pragma unroll
    for (int r = 0; r < 8; ++r) cp[(size_t)r * CH] = acc[r];
}

// ---------------------------------------------------------------------------
// Host-side launch
// ---------------------------------------------------------------------------
extern "C" void kernel_launch(void* const* d_in, const int* in_sizes, int n_in,
                              void* d_out, int out_size, void* d_ws, size_t ws_size,
                              hipStream_t stream) {
    const float*     x   = (const float*)d_in[0];
    const long long* ei  = (const long long*)d_in[1];   // int64 edge_index [2, E]
    const float*     W1  = (const float*)d_in[2];
    const float*     b1  = (const float*)d_in[3];
    const float*     W2  = (const float*)d_in[4];
    const float*     b2  = (const float*)d_in[5];

    const long long N = in_sizes[0] / 256;              // 100000
    const long long E = in_sizes[1] / 2;                // 1600000
    const long long NC = N * CH;                        // 6.4M feature elements

    const long long* src = ei;
    const long long* dst = ei + E;

    // Workspace: dinv[N] + bufA[N*64] + bufB[N*64]  (~52 MB)
    char* wsb = (char*)d_ws;
    size_t off = 0;
    auto carve = [&](size_t bytes) -> float* {
        float* p = (float*)(wsb + off);
        off = (off + bytes + 255) & ~(size_t)255;
        return p;
    };
    float* dinv = carve((size_t)N * 4);
    float* bufA = carve((size_t)NC * 4);
    float* bufB = carve((size_t)NC * 4);

    float* out_graph   = (float*)d_out;                 // [N,64]
    float* out_nograph = (float*)d_out + NC;            // [N,64]

    const int TPB = 256;
    dim3 blk(TPB);
    int gN   = (int)((N  + TPB - 1) / TPB);
    int gNC  = (int)((NC + TPB - 1) / TPB);
    int gE   = (int)((E  + TPB - 1) / TPB);
    int gE16 = (int)((E * 16 + TPB - 1) / TPB);
    int nTiles = (int)((N / 16) * (CH / 16));
    int gGemm  = (nTiles + WAVES_PER_BLOCK - 1) / WAVES_PER_BLOCK;

    // --- degrees: deg = 1 (self loop) + scatter(ones, dst); dinv = rsqrt(deg)
    fill_f32<<<gN, blk, 0, stream>>>(dinv, 1.0f, N);
    deg_accum<<<gE, blk, 0, stream>>>(dst, dinv, E);
    rsqrt_inplace<<<gN, blk, 0, stream>>>(dinv, N);

    // --- layer 1: h = x @ W1 ; agg = scatter + self ; h1 = relu(agg + b1)
    gemm_wmma_f32<256><<<gGemm, blk, 0, stream>>>(x, W1, bufA, (int)N);
    self_loop_init<<<gNC, blk, 0, stream>>>(bufA, dinv, bufB, NC);
    edge_scatter<<<gE16, blk, 0, stream>>>(src, dst, dinv, bufA, bufB, E);
    bias_act_inplace<<<gNC, blk, 0, stream>>>(bufB, b1, NC, /*relu=*/1);

    // --- layer 2 GEMM: h2 = h1 @ W2  (bufB -> bufA)
    gemm_wmma_f32<64><<<gGemm, blk, 0, stream>>>(bufB, W2, bufA, (int)N);

    // --- out_nograph = h2 + b2
    add_bias_store<<<gNC, blk, 0, stream>>>(bufA, b2, out_nograph, NC);

    // --- out_graph = scatter(h2) + self + b2, accumulated directly in d_out
    self_loop_init<<<gNC, blk, 0, stream>>>(bufA, dinv, out_graph, NC);
    edge_scatter<<<gE16, blk, 0, stream>>>(src, dst, dinv, bufA, out_graph, E);
    bias_act_inplace<<<gNC, blk, 0, stream>>>(out_graph, b2, NC, /*relu=*/0);
}